// Linear_54271206752526
// MI455X (gfx1250) — compile-verified
//
#include <hip/hip_runtime.h>

#define BATCH 256
#define FIN   1024
#define FOUT  1024

typedef float v2f __attribute__((ext_vector_type(2)));
typedef float v4f __attribute__((ext_vector_type(4)));
typedef float v8f __attribute__((ext_vector_type(8)));

__device__ __forceinline__ float softplus_f(float x) {
    // numerically stable log(1 + exp(x))
    return fmaxf(x, 0.0f) + log1pf(expf(-fabsf(x)));
}

// ---------------------------------------------------------------------------
// Kernel 1: sigma_w = softplus(weight_rho)   (1M elems -> d_ws, 4 MB)
// Runs once; keeps transcendentals out of the 268M-element streaming loop.
// ---------------------------------------------------------------------------
__global__ void __launch_bounds__(256) sigma_kernel(const float* __restrict__ rho,
                                                    float* __restrict__ sigma) {
    int i = blockIdx.x * 256 + threadIdx.x;
    sigma[i] = softplus_f(rho[i]);
}

// ---------------------------------------------------------------------------
// Kernel 2: Y = X @ MU^T via V_WMMA_F32_16X16X4_F32.
// One wave per 16x16 output tile; K loop in steps of 4.
// A lane layout (16x4 f32 A): lane l -> M = l&15, K = k0 + 2*(l>>4) .. +1
// B lane layout (4x16 f32 B): B[k][n] = MU[n][k] -> same float2 pattern on MU row.
// C/D layout: VGPR v -> row v + 8*(l>>4), col l&15.
// ---------------------------------------------------------------------------
__global__ void __launch_bounds__(128) gemm_mu_kernel(const float* __restrict__ X,
                                                      const float* __restrict__ MU,
                                                      float* __restrict__ Y) {
    const int wave = threadIdx.x >> 5;
    const int lane = threadIdx.x & 31;
    const int tile = blockIdx.x * 4 + wave;      // 1024 tiles total
    const int tm   = tile >> 6;                  // 16 b-tiles
    const int tn   = tile & 63;                  // 64 o-tiles
    const int m    = lane & 15;
    const int hsel = lane >> 4;
    const int kk   = hsel * 2;

    const float* xrow = X  + (size_t)(tm * 16 + m) * FIN;
    const float* wrow = MU + (size_t)(tn * 16 + m) * FIN;

    v8f c = {};
    #pragma unroll 4
    for (int k0 = 0; k0 < FIN; k0 += 4) {
        v2f a = *(const v2f*)(xrow + k0 + kk);
        v2f b = *(const v2f*)(wrow + k0 + kk);
        c = __builtin_amdgcn_wmma_f32_16x16x4_f32(
                /*neg_a=*/false, a, /*neg_b=*/false, b,
                /*c_mod=*/(short)0, c, /*reuse_a=*/false, /*reuse_b=*/false);
    }

    const int row = tm * 16 + hsel * 8;
    const int col = tn * 16 + m;
    #pragma unroll
    for (int v = 0; v < 8; ++v) {
        Y[(size_t)(row + v) * FOUT + col] = c[v];
    }
}

// ---------------------------------------------------------------------------
// Kernel 3: streaming reduction of the stochastic term (HBM-bound, 1 GiB).
// Block = 256 threads = 8 waves; block handles (b, 8 consecutive o).
// x-row staged in LDS once per block; sigma row from L2 (regular temporal
// hint -> stays resident); eps_w streamed exactly once with NON-TEMPORAL
// b128 loads so the 1 GiB use-once stream doesn't evict the reused 9 MB
// working set (x, sigma, mu) from L2. Wave32 butterfly reduction; lane 0
// adds bias and accumulates into Y (written by kernel 2).
// ---------------------------------------------------------------------------
__global__ void __launch_bounds__(256) eps_kernel(const float* __restrict__ X,
                                                  const float* __restrict__ SIG,
                                                  const float* __restrict__ bias_mu,
                                                  const float* __restrict__ bias_rho,
                                                  const float* __restrict__ EPSW,
                                                  const float* __restrict__ EPSB,
                                                  float* __restrict__ Y) {
    __shared__ float xs[FIN];
    const int b  = blockIdx.x >> 7;     // / (FOUT/8)
    const int og = blockIdx.x & 127;
    const int t  = threadIdx.x;

    // stage x[b, :] : 256 threads x float4 = 1024 floats
    ((v4f*)xs)[t] = ((const v4f*)(X + (size_t)b * FIN))[t];
    __syncthreads();

    const int wave = t >> 5;
    const int lane = t & 31;
    const int o    = og * 8 + wave;

    const float* sig = SIG  + (size_t)o * FIN;
    const float* ep  = EPSW + ((size_t)b * FOUT + o) * FIN;

    float acc = 0.0f;
    #pragma unroll
    for (int k = 0; k < 8; ++k) {
        const int idx = k * 128 + lane * 4;
        v4f xv = *(const v4f*)(xs + idx);
        v4f sv = *(const v4f*)(sig + idx);
        v4f ev = __builtin_nontemporal_load((const v4f*)(ep + idx));
        acc = fmaf(xv.x * sv.x, ev.x, acc);
        acc = fmaf(xv.y * sv.y, ev.y, acc);
        acc = fmaf(xv.z * sv.z, ev.z, acc);
        acc = fmaf(xv.w * sv.w, ev.w, acc);
    }

    // wave32 butterfly reduction
    #pragma unroll
    for (int off = 16; off > 0; off >>= 1)
        acc += __shfl_xor(acc, off, 32);

    if (lane == 0) {
        const float eb = __builtin_nontemporal_load(EPSB + (size_t)b * FOUT + o);
        const float bias = bias_mu[o] + softplus_f(bias_rho[o]) * eb;
        Y[(size_t)b * FOUT + o] += acc + bias;
    }
}

extern "C" void kernel_launch(void* const* d_in, const int* in_sizes, int n_in,
                              void* d_out, int out_size, void* d_ws, size_t ws_size,
                              hipStream_t stream) {
    const float* x    = (const float*)d_in[0];
    const float* wmu  = (const float*)d_in[1];
    const float* wrho = (const float*)d_in[2];
    const float* bmu  = (const float*)d_in[3];
    const float* brho = (const float*)d_in[4];
    const float* epsw = (const float*)d_in[5];
    const float* epsb = (const float*)d_in[6];
    float* out   = (float*)d_out;
    float* sigma = (float*)d_ws;          // FOUT*FIN floats = 4 MB scratch

    // 1) sigma = softplus(rho)
    sigma_kernel<<<(FOUT * FIN) / 256, 256, 0, stream>>>(wrho, sigma);
    // 2) out = x @ mu^T  (WMMA f32)
    gemm_mu_kernel<<<256, 128, 0, stream>>>(x, wmu, out);
    // 3) out += sum_i x*sigma*eps_w  + bias
    eps_kernel<<<BATCH * (FOUT / 8), 256, 0, stream>>>(x, sigma, bmu, brho,
                                                       epsw, epsb, out);
}